// CausalSelfAttention_31533649888027
// MI455X (gfx1250) — compile-verified
//
#include <hip/hip_runtime.h>
#include <hip/hip_bf16.h>

#define D_MODEL  512
#define N_HEADS  8
#define HEAD_DIM 64
#define B_SZ     4
#define L_SEQ    2048
#define HPAD     16   // padded head count (rows/cols 8..15 are exact zeros)

typedef __attribute__((ext_vector_type(2))) float v2f;
typedef __attribute__((ext_vector_type(4))) float v4f;
typedef __attribute__((ext_vector_type(8))) float v8f;

// Workspace layout (in floats):
//   Upad   : [B][16][512]   @ 0        (32768)   rows 8..15 = 0
//   scores : [B][16][2048]  @ 32768    (131072)  rows 8..15 = 0; softmaxed in place
//   c      : [B][8][512]    @ 163840   (16384)
#define WS_U 0
#define WS_S 32768
#define WS_C 163840

// ---------------------------------------------------------------------------
// k1: per-batch edge math.
//   q_last = Wq @ h_last ; g = (tau/8)*q_last + delta_last
//   Upad[b][head][:] = Wk_head^T @ g_head  (head < 8), rows 8..15 zeroed.
// ---------------------------------------------------------------------------
__global__ void k1_project_u(const float* __restrict__ h,
                             const float* __restrict__ tau,
                             const float* __restrict__ delta,
                             const float* __restrict__ Wq,
                             const float* __restrict__ Wk,
                             float* __restrict__ Upad) {
  __shared__ float s_h[D_MODEL];
  __shared__ float s_g[D_MODEL];
  const int b   = blockIdx.x;
  const int tid = threadIdx.x;

  const float* hlast = h + ((size_t)b * L_SEQ + (L_SEQ - 1)) * D_MODEL;
  for (int d = tid; d < D_MODEL; d += blockDim.x) s_h[d] = hlast[d];
  __syncthreads();

  const float tscale = tau[b * L_SEQ + (L_SEQ - 1)] * 0.125f;  // tau / sqrt(64)
  const float* dlast = delta + ((size_t)b * L_SEQ + (L_SEQ - 1)) * D_MODEL;
  for (int i = tid; i < D_MODEL; i += blockDim.x) {
    const float* wrow = Wq + (size_t)i * D_MODEL;
    float acc = 0.f;
    for (int j = 0; j < D_MODEL; ++j) acc += wrow[j] * s_h[j];
    s_g[i] = tscale * acc + dlast[i];
  }
  __syncthreads();

  float* U = Upad + (size_t)b * HPAD * D_MODEL;
  for (int idx = tid; idx < HPAD * D_MODEL; idx += blockDim.x) {
    const int head = idx >> 9;            // /512
    const int d    = idx & (D_MODEL - 1);
    float acc = 0.f;
    if (head < N_HEADS) {
      const float* g = s_g + head * HEAD_DIM;
      for (int j = 0; j < HEAD_DIM; ++j)
        acc += g[j] * Wk[((size_t)(head * HEAD_DIM + j)) * D_MODEL + d];
    }
    U[idx] = acc;
  }
}

// ---------------------------------------------------------------------------
// k2: score pass, WMMA f32 16x16x4, branch-free inner loop.
//   Per wave: 16 key rows x 16 head cols (8..15 zero), K over D_MODEL.
//   b128 loads cover 2 WMMA steps (K permuted consistently for A and B).
// ---------------------------------------------------------------------------
__global__ void k2_scores(const float* __restrict__ h,
                          const float* __restrict__ Upad,
                          float* __restrict__ scores) {
  const int wave  = threadIdx.x >> 5;
  const int lane  = threadIdx.x & 31;
  const int gtile = blockIdx.x * (blockDim.x >> 5) + wave;  // 0 .. B*L/16-1
  const int b     = gtile >> 7;                             // 128 tiles / batch
  const int k0    = (gtile & 127) << 4;
  const int hl    = lane >> 4;   // lane half
  const int m     = lane & 15;   // A: key row; B: head col

  const float* hrow = h    + ((size_t)b * L_SEQ + k0 + m) * D_MODEL + 4 * hl;
  const float* urow = Upad + ((size_t)b * HPAD + m) * D_MODEL + 4 * hl;

  v8f acc = {};
#pragma unroll 4
  for (int kk = 0; kk < D_MODEL; kk += 8) {
    v4f av = *(const v4f*)(hrow + kk);
    v4f bv = *(const v4f*)(urow + kk);
    v2f a0 = {av.x, av.y}, b0 = {bv.x, bv.y};
    acc = __builtin_amdgcn_wmma_f32_16x16x4_f32(false, a0, false, b0,
                                                (short)0, acc, false, false);
    v2f a1 = {av.z, av.w}, b1 = {bv.z, bv.w};
    acc = __builtin_amdgcn_wmma_f32_16x16x4_f32(false, a1, false, b1,
                                                (short)0, acc, false, false);
  }

  // D: row M = r + 8*hl (key pos), col N = m (head). Cols 8..15 are exact 0.
  float* srow = scores + ((size_t)b * HPAD + m) * L_SEQ + k0;
#pragma unroll
  for (int r = 0; r < 8; ++r) {
    float v = acc[r];
    v = fminf(fmaxf(v, -50.f), 50.f);
    srow[r + 8 * hl] = v;
  }
}

// ---------------------------------------------------------------------------
// k3: softmax over L=2048, one block per real (b,head) row, in place.
// ---------------------------------------------------------------------------
__global__ void k3_softmax(float* __restrict__ scores) {
  __shared__ float red[256];
  const int b    = blockIdx.x >> 3;
  const int head = blockIdx.x & 7;
  float* s       = scores + ((size_t)b * HPAD + head) * L_SEQ;
  const int tid  = threadIdx.x;

  float m = -1e30f;
  for (int k = tid; k < L_SEQ; k += blockDim.x) m = fmaxf(m, s[k]);
  red[tid] = m; __syncthreads();
  for (int off = 128; off > 0; off >>= 1) {
    if (tid < off) red[tid] = fmaxf(red[tid], red[tid + off]);
    __syncthreads();
  }
  m = red[0]; __syncthreads();

  float sum = 0.f;
  for (int k = tid; k < L_SEQ; k += blockDim.x) {
    float e = expf(s[k] - m);
    s[k] = e;
    sum += e;
  }
  red[tid] = sum; __syncthreads();
  for (int off = 128; off > 0; off >>= 1) {
    if (tid < off) red[tid] += red[tid + off];
    __syncthreads();
  }
  const float inv = 1.f / red[0];
  for (int k = tid; k < L_SEQ; k += blockDim.x) s[k] *= inv;
}

// ---------------------------------------------------------------------------
// k4: weighted-sum pass, WMMA f32 16x16x4, branch-free inner loop.
//   c[b,h,:] = sum_k w[b,h,k] * h[b,k,:]
//   A = weights [16 head rows, 8..15 zero], B = h rows; K over L.
// ---------------------------------------------------------------------------
__global__ void k4_wsum(const float* __restrict__ h,
                        const float* __restrict__ w,
                        float* __restrict__ c) {
  const int wave  = threadIdx.x >> 5;
  const int lane  = threadIdx.x & 31;
  const int gtile = blockIdx.x * (blockDim.x >> 5) + wave;  // 0 .. B*32-1
  const int b     = gtile >> 5;
  const int n0    = (gtile & 31) << 4;
  const int hl    = lane >> 4;
  const int q     = lane & 15;   // A: head row; B: dim col

  const float* wrow = w + ((size_t)b * HPAD + q) * L_SEQ + 4 * hl;
  const float* hb   = h + (size_t)b * L_SEQ * D_MODEL + n0 + q;

  v8f acc = {};
#pragma unroll 2
  for (int k0 = 0; k0 < L_SEQ; k0 += 8) {
    v4f av = *(const v4f*)(wrow + k0);
    const size_t kbase = (size_t)(k0 + 4 * hl);
    float b00 = hb[kbase * D_MODEL];
    float b01 = hb[(kbase + 1) * D_MODEL];
    float b10 = hb[(kbase + 2) * D_MODEL];
    float b11 = hb[(kbase + 3) * D_MODEL];
    v2f a0 = {av.x, av.y}, bb0 = {b00, b01};
    acc = __builtin_amdgcn_wmma_f32_16x16x4_f32(false, a0, false, bb0,
                                                (short)0, acc, false, false);
    v2f a1 = {av.z, av.w}, bb1 = {b10, b11};
    acc = __builtin_amdgcn_wmma_f32_16x16x4_f32(false, a1, false, bb1,
                                                (short)0, acc, false, false);
  }

  // D row M = r + 8*hl (head). hl==1 rows are the zero pad heads -> drop.
  if (hl == 0) {
#pragma unroll
    for (int r = 0; r < 8; ++r)
      c[((size_t)b * N_HEADS + r) * D_MODEL + n0 + q] = acc[r];
  }
}

// ---------------------------------------------------------------------------
// k5: per batch. attn[j] = dot(Wv[j,:], c[b, j/64, :]); out = Wo@attn + bo.
// ---------------------------------------------------------------------------
__global__ void k5_out(const float* __restrict__ c,
                       const float* __restrict__ Wv,
                       const float* __restrict__ Wo,
                       const float* __restrict__ bo,
                       float* __restrict__ out) {
  __shared__ float s_attn[D_MODEL];
  const int b   = blockIdx.x;
  const int tid = threadIdx.x;
  const float* cb = c + (size_t)b * N_HEADS * D_MODEL;

  for (int j = tid; j < D_MODEL; j += blockDim.x) {
    const float* ch = cb + (j >> 6) * D_MODEL;   // head = j/64
    const float* wv = Wv + (size_t)j * D_MODEL;
    float acc = 0.f;
    for (int d = 0; d < D_MODEL; ++d) acc += wv[d] * ch[d];
    s_attn[j] = acc;
  }
  __syncthreads();

  for (int i = tid; i < D_MODEL; i += blockDim.x) {
    const float* wo = Wo + (size_t)i * D_MODEL;
    float acc = bo[i];
    for (int d = 0; d < D_MODEL; ++d) acc += wo[d] * s_attn[d];
    out[b * D_MODEL + i] = acc;
  }
}

// ---------------------------------------------------------------------------
extern "C" void kernel_launch(void* const* d_in, const int* in_sizes, int n_in,
                              void* d_out, int out_size, void* d_ws, size_t ws_size,
                              hipStream_t stream) {
  const float* h     = (const float*)d_in[0];
  const float* tau   = (const float*)d_in[1];
  const float* delta = (const float*)d_in[2];
  const float* Wq    = (const float*)d_in[3];
  const float* Wk    = (const float*)d_in[4];
  const float* Wv    = (const float*)d_in[5];
  const float* Wo    = (const float*)d_in[6];
  const float* bo    = (const float*)d_in[7];
  float* out = (float*)d_out;
  float* ws  = (float*)d_ws;

  float* Upad   = ws + WS_U;
  float* scores = ws + WS_S;   // padded [B][16][L]; softmaxed in place
  float* cbuf   = ws + WS_C;

  // 1) edge math -> padded U
  k1_project_u<<<B_SZ, 256, 0, stream>>>(h, tau, delta, Wq, Wk, Upad);
  // 2) score pass: B*L/16 = 512 waves -> 64 blocks x 8 waves
  k2_scores<<<(B_SZ * L_SEQ / 16) / 8, 256, 0, stream>>>(h, Upad, scores);
  // 3) softmax: 32 real rows
  k3_softmax<<<B_SZ * N_HEADS, 256, 0, stream>>>(scores);
  // 4) weighted sum: B*32 = 128 waves -> 16 blocks x 8 waves
  k4_wsum<<<(B_SZ * 32) / 8, 256, 0, stream>>>(h, scores, cbuf);
  // 5) Wv / Wo / bo epilogue
  k5_out<<<B_SZ, 256, 0, stream>>>(cbuf, Wv, Wo, bo, out);
}